// Seg_Loss_74337293959545
// MI455X (gfx1250) — compile-verified
//
#include <hip/hip_runtime.h>
#include <hip/hip_bf16.h>
#include <math.h>

typedef __attribute__((ext_vector_type(2))) float v2f;
typedef __attribute__((ext_vector_type(8))) float v8f;

#define BDIM 256
#define GDIM 1024

constexpr int  Bn  = 4, Cn = 3, NCn = 8;
constexpr long HW   = 1024L * 1024L;       // H*W = 2^20
constexpr long Npix = (long)Bn * HW;       // 2^22
constexpr float invN = 1.0f / (float)Npix;

// ---------------------------------------------------------------------------
// ws layout: floats [0..23] = {Sx,Sy,cnt} per label L at [L*3+{0,1,2}]
//            floats [24],[25] = s1, s3 partials
//            (optional, if ws_size allows) at float offset 64:
//              wsx[Npix] f32, wsy[Npix] f32, wslab[Npix] u8
// ---------------------------------------------------------------------------

__global__ void seg_init_ws(float* ws) {
  int t = blockIdx.x * blockDim.x + threadIdx.x;
  if (t < 32) ws[t] = 0.0f;
}

__device__ __forceinline__ float fcomp(float4 v, int q) {
  return q == 0 ? v.x : q == 1 ? v.y : q == 2 ? v.z : v.w;   // folds when q const
}

// Load a 4-pixel group (b128 per channel) and produce x_avg/y_avg/argmax-label.
__device__ __forceinline__ void load_group(const float* __restrict__ x,
                                           const float* __restrict__ y,
                                           const float* __restrict__ mask,
                                           long b, long s,
                                           float xa[4], float ya[4], int lab[4]) {
  float4 mv[8];
#pragma unroll
  for (int c = 0; c < NCn; ++c)
    mv[c] = *(const float4*)(mask + (b * NCn + c) * HW + s);
  float4 xv[3], yv[3];
#pragma unroll
  for (int c = 0; c < Cn; ++c) {
    xv[c] = *(const float4*)(x + (b * Cn + c) * HW + s);
    yv[c] = *(const float4*)(y + (b * Cn + c) * HW + s);
  }
#pragma unroll
  for (int q = 0; q < 4; ++q) {
    float best = fcomp(mv[0], q);
    int l = 0;
#pragma unroll
    for (int c = 1; c < NCn; ++c) {          // first-max tie-break == jnp.argmax
      float v = fcomp(mv[c], q);
      if (v > best) { best = v; l = c; }
    }
    lab[q] = l;
    xa[q] = (fcomp(xv[0], q) + fcomp(xv[1], q) + fcomp(xv[2], q)) * (1.0f / 3.0f);
    ya[q] = (fcomp(yv[0], q) + fcomp(yv[1], q) + fcomp(yv[2], q)) * (1.0f / 3.0f);
  }
}

// Pass 1: segmented per-label reduction on the matrix unit.
// D += A(one-hot labels, 16x4) x B([x,y,1], 4x16) via V_WMMA_F32_16X16X4_F32.
template <bool SAVE>
__global__ __launch_bounds__(BDIM) void seg_pass1(const float* __restrict__ x,
                                                  const float* __restrict__ y,
                                                  const float* __restrict__ mask,
                                                  float* __restrict__ gbins,
                                                  float* __restrict__ wsx,
                                                  float* __restrict__ wsy,
                                                  unsigned char* __restrict__ wslab) {
  __shared__ float sbins[24];
  if (threadIdx.x < 24) sbins[threadIdx.x] = 0.0f;
  __syncthreads();

  const int lane  = threadIdx.x & 31;
  const int kbase = (lane >> 4) << 1;   // VGPR0 holds K=0 (lanes 0-15) / K=2 (16-31)
  const int col   = lane & 15;          // B/D column
  const int m     = lane & 15;          // A row (label)
  v8f acc = {};

  const long T4 = (long)GDIM * BDIM * 4;              // pixels per sweep (2^20)
  long p = ((long)blockIdx.x * BDIM + threadIdx.x) * 4;
  const int iters = (int)(Npix / T4);                 // 4; exact, EXEC stays full

  for (int it = 0; it < iters; ++it, p += T4) {
    const long b = p >> 20, s = p & (HW - 1);

    if (it + 1 < iters) {               // gfx1250 global_prefetch_b8
      const long pn = p + T4, bn2 = pn >> 20, sn = pn & (HW - 1);
      __builtin_prefetch(mask + (bn2 * NCn) * HW + sn, 0, 1);
      __builtin_prefetch(x + (bn2 * Cn) * HW + sn, 0, 1);
      __builtin_prefetch(y + (bn2 * Cn) * HW + sn, 0, 1);
    }

    float xa[4], ya[4];
    int lab[4];
    load_group(x, y, mask, b, s, xa, ya, lab);

    if (SAVE) {                         // spill intermediates for pass 2 (b128)
      *(float4*)(wsx + p) = make_float4(xa[0], xa[1], xa[2], xa[3]);
      *(float4*)(wsy + p) = make_float4(ya[0], ya[1], ya[2], ya[3]);
      *(uchar4*)(wslab + p) =
          make_uchar4((unsigned char)lab[0], (unsigned char)lab[1],
                      (unsigned char)lab[2], (unsigned char)lab[3]);
    }

#pragma unroll
    for (int q = 0; q < 4; ++q) {       // one 8-WMMA binning round per pixel-quad
#pragma unroll
      for (int j = 0; j < 8; ++j) {
        const int p0 = j * 4 + kbase, p1 = p0 + 1;
        int   l0 = __shfl(lab[q], p0, 32);
        int   l1 = __shfl(lab[q], p1, 32);
        float x0 = __shfl(xa[q], p0, 32);
        float x1 = __shfl(xa[q], p1, 32);
        float y0 = __shfl(ya[q], p0, 32);
        float y1 = __shfl(ya[q], p1, 32);
        v2f A, Bm;
        A.x  = (l0 == m) ? 1.0f : 0.0f;
        A.y  = (l1 == m) ? 1.0f : 0.0f;
        Bm.x = (col == 0) ? x0 : (col == 1) ? y0 : (col == 2) ? 1.0f : 0.0f;
        Bm.y = (col == 0) ? x1 : (col == 1) ? y1 : (col == 2) ? 1.0f : 0.0f;
        acc = __builtin_amdgcn_wmma_f32_16x16x4_f32(false, A, false, Bm,
                                                    (short)0, acc, false, false);
      }
    }
  }

  // D layout: VGPR r = rows M=r (lanes 0-15) / M=r+8 (lanes 16-31), N=lane.
  // Bins at rows L=0..7, cols 0..2 -> lanes 0..2, VGPRs 0..7.
  if (lane < 3) {
#pragma unroll
    for (int L = 0; L < 8; ++L) atomicAdd(&sbins[L * 3 + lane], acc[L]);
  }
  __syncthreads();
  if (threadIdx.x < 24) atomicAdd(&gbins[threadIdx.x], sbins[threadIdx.x]);
}

// Pass 2: per-pixel pow/variance terms against per-label means.
template <bool USEWS>
__global__ __launch_bounds__(BDIM) void seg_pass2(const float* __restrict__ x,
                                                  const float* __restrict__ y,
                                                  const float* __restrict__ mask,
                                                  const float* __restrict__ wsx,
                                                  const float* __restrict__ wsy,
                                                  const unsigned char* __restrict__ wslab,
                                                  const float* __restrict__ gbins,
                                                  float* __restrict__ gacc) {
  __shared__ float sxbar[8], sybar[8], spart[2];
  if (threadIdx.x < 8) {
    sxbar[threadIdx.x] = gbins[threadIdx.x * 3 + 0] * invN;
    sybar[threadIdx.x] = gbins[threadIdx.x * 3 + 1] * invN;
  }
  if (threadIdx.x < 2) spart[threadIdx.x] = 0.0f;
  __syncthreads();

  float s1 = 0.0f, s3 = 0.0f;
  const long T4 = (long)GDIM * BDIM * 4;
  long p = ((long)blockIdx.x * BDIM + threadIdx.x) * 4;
  const int iters = (int)(Npix / T4);

  for (int it = 0; it < iters; ++it, p += T4) {
    float xa[4], ya[4];
    int lab[4];
    if (USEWS) {
      if (it + 1 < iters) {
        __builtin_prefetch(wsx + p + T4, 0, 1);
        __builtin_prefetch(wsy + p + T4, 0, 1);
        __builtin_prefetch(wslab + p + T4, 0, 1);
      }
      float4 xv = *(const float4*)(wsx + p);
      float4 yv = *(const float4*)(wsy + p);
      uchar4 lv = *(const uchar4*)(wslab + p);
      xa[0] = xv.x; xa[1] = xv.y; xa[2] = xv.z; xa[3] = xv.w;
      ya[0] = yv.x; ya[1] = yv.y; ya[2] = yv.z; ya[3] = yv.w;
      lab[0] = lv.x; lab[1] = lv.y; lab[2] = lv.z; lab[3] = lv.w;
    } else {
      const long b = p >> 20, s = p & (HW - 1);
      if (it + 1 < iters) {
        const long pn = p + T4, bn2 = pn >> 20, sn = pn & (HW - 1);
        __builtin_prefetch(mask + (bn2 * NCn) * HW + sn, 0, 1);
        __builtin_prefetch(x + (bn2 * Cn) * HW + sn, 0, 1);
        __builtin_prefetch(y + (bn2 * Cn) * HW + sn, 0, 1);
      }
      load_group(x, y, mask, b, s, xa, ya, lab);
    }

#pragma unroll
    for (int q = 0; q < 4; ++q) {
      float xb = sxbar[lab[q]];         // LDS-indexed per-label means
      float yb = sybar[lab[q]];
      float xr = (xa[q] != 0.0f) ? xa[q] : xb;
      float yr = (ya[q] != 0.0f) ? ya[q] : yb;
      s1 += powf(xr, yr);
      float d = xr - xb;
      s3 += d * d;
    }
  }

  for (int off = 16; off > 0; off >>= 1) {
    s1 += __shfl_down(s1, off, 32);
    s3 += __shfl_down(s3, off, 32);
  }
  if ((threadIdx.x & 31) == 0) {
    atomicAdd(&spart[0], s1);
    atomicAdd(&spart[1], s3);
  }
  __syncthreads();
  if (threadIdx.x == 0) {
    atomicAdd(&gacc[0], spart[0]);
    atomicAdd(&gacc[1], spart[1]);
  }
}

__global__ void seg_finalize(const float* __restrict__ ws, float* __restrict__ out) {
  if (threadIdx.x != 0 || blockIdx.x != 0) return;
  const float s1 = ws[24], s3 = ws[25];
  float l2 = 0.0f, extra = 0.0f;
#pragma unroll
  for (int L = 0; L < 8; ++L) {
    float Sx = ws[L * 3 + 0], Sy = ws[L * 3 + 1], cnt = ws[L * 3 + 2];
    float xb = Sx * invN, yb = Sy * invN;
    float P = powf(xb, yb);             // powf(0,0)==1 matches jnp for empty labels
    l2 += P;
    extra += ((float)Npix - cnt) * P;   // out-of-region pixels each contribute P
  }
  out[0] = (s1 + extra) * invN;         // loss1
  out[1] = l2;                          // loss2
  out[2] = s3 * invN;                   // loss3
}

extern "C" void kernel_launch(void* const* d_in, const int* in_sizes, int n_in,
                              void* d_out, int out_size, void* d_ws, size_t ws_size,
                              hipStream_t stream) {
  const float* x    = (const float*)d_in[0];
  const float* y    = (const float*)d_in[1];
  const float* mask = (const float*)d_in[2];
  float* ws  = (float*)d_ws;
  float* out = (float*)d_out;

  float* wsx = ws + 64;                        // 256 B in
  float* wsy = wsx + Npix;
  unsigned char* wslab = (unsigned char*)(wsy + Npix);
  const size_t need = 256 + (size_t)Npix * 9;  // 37.75 MB
  const bool save = ws_size >= need;           // fixed per run -> deterministic

  seg_init_ws<<<1, 64, 0, stream>>>(ws);
  if (save) {
    seg_pass1<true><<<GDIM, BDIM, 0, stream>>>(x, y, mask, ws, wsx, wsy, wslab);
    seg_pass2<true><<<GDIM, BDIM, 0, stream>>>(x, y, mask, wsx, wsy, wslab, ws, ws + 24);
  } else {
    seg_pass1<false><<<GDIM, BDIM, 0, stream>>>(x, y, mask, ws, wsx, wsy, wslab);
    seg_pass2<false><<<GDIM, BDIM, 0, stream>>>(x, y, mask, wsx, wsy, wslab, ws, ws + 24);
  }
  seg_finalize<<<1, 32, 0, stream>>>(ws, out);
}